// PrototypicalLoss_64536178589961
// MI455X (gfx1250) — compile-verified
//
#include <hip/hip_runtime.h>
#include <hip/hip_bf16.h>

typedef __attribute__((ext_vector_type(2))) float v2f;
typedef __attribute__((ext_vector_type(8))) float v8f;

#define NROWS 131072
#define DIM   512
#define NCLS  10
#define PSTRIDE 516   // LDS row stride for prototypes: 516 % 64 == 4 -> conflict-free b64 loads

// workspace layout (float offsets)
#define OFF_SUMS   0        // 10*512
#define OFF_COUNTS 5120     // 16
#define OFF_LOSS   5136     // 1
#define OFF_P      6144     // 16*512 (rows 10..15 zero)
#define OFF_SQP    14336    // 16
#define WS_FLOATS  14352

// ---------- wave helpers (wave32) ----------
template <int MASK>
__device__ __forceinline__ float swz_xor(float v) {
  // group-of-32 swizzle: and=0x1f, or=0, xor=MASK
  return __int_as_float(
      __builtin_amdgcn_ds_swizzle(__float_as_int(v), (MASK << 10) | 0x1f));
}
__device__ __forceinline__ float lane_bcast(float v, int srcLane) {
  return __int_as_float(
      __builtin_amdgcn_ds_bpermute(srcLane << 2, __float_as_int(v)));
}

// ---------- kernel 0: zero workspace accumulators ----------
__global__ void k_init(float* __restrict__ ws) {
  int i = blockIdx.x * 256 + threadIdx.x;
  if (i < WS_FLOATS) ws[i] = 0.0f;
}

// ---------- kernel 1: per-class sums & counts ----------
__global__ __launch_bounds__(256) void k_protosum(const float* __restrict__ E,
                                                  const int* __restrict__ lab,
                                                  float* __restrict__ ws) {
  __shared__ float ls[NCLS * DIM];
  __shared__ float lc[NCLS];
  const int t = threadIdx.x;
  for (int i = t; i < NCLS * DIM; i += 256) ls[i] = 0.0f;
  if (t < NCLS) lc[t] = 0.0f;
  __syncthreads();

  const int r0 = blockIdx.x * 256;
  for (int r = r0; r < r0 + 256; ++r) {
    const int c = lab[r];                       // uniform -> scalar load
    const float* row = E + (size_t)r * DIM;
    // thread t owns dims {t, t+256}: no cross-thread LDS conflicts, no syncs
    ls[c * DIM + t]       += row[t];
    ls[c * DIM + t + 256] += row[t + 256];
    if (t == 0) lc[c] += 1.0f;
  }
  __syncthreads();
  for (int i = t; i < NCLS * DIM; i += 256) atomicAdd(&ws[OFF_SUMS + i], ls[i]);
  if (t < NCLS) atomicAdd(&ws[OFF_COUNTS + t], lc[t]);
}

// ---------- kernel 2: prototypes (zero-padded to 16 rows) + ||p||^2 ----------
__global__ __launch_bounds__(256) void k_proto(float* __restrict__ ws) {
  __shared__ float red[256];
  const int c = blockIdx.x;   // 0..15
  const int t = threadIdx.x;
  float p0 = 0.0f, p1 = 0.0f;
  if (c < NCLS) {
    const float inv = 1.0f / ws[OFF_COUNTS + c];
    p0 = ws[OFF_SUMS + c * DIM + t] * inv;
    p1 = ws[OFF_SUMS + c * DIM + t + 256] * inv;
  }
  ws[OFF_P + c * DIM + t]       = p0;
  ws[OFF_P + c * DIM + t + 256] = p1;
  red[t] = p0 * p0 + p1 * p1;
  __syncthreads();
  for (int s = 128; s > 0; s >>= 1) {
    if (t < s) red[t] += red[t + s];
    __syncthreads();
  }
  if (t == 0) ws[OFF_SQP + c] = red[0];
}

// ---------- kernel 3: WMMA distance GEMM + log-softmax + loss ----------
__global__ __launch_bounds__(256) void k_logits(const float* __restrict__ E,
                                                const int* __restrict__ lab,
                                                const float* __restrict__ ws,
                                                float* __restrict__ logits,
                                                float* __restrict__ loss_acc) {
  __shared__ float lp[16 * PSTRIDE];
  __shared__ float lsqp[16];
  __shared__ float lloss[8];
  const int t = threadIdx.x;

  // stage padded prototype matrix into LDS (conflict-free stride 516)
  for (int i = t; i < 16 * DIM; i += 256) {
    const int rr = i >> 9;
    const int cc = i & 511;
    lp[rr * PSTRIDE + cc] = ws[OFF_P + i];
  }
  if (t < 16) lsqp[t] = ws[OFF_SQP + t];
  __syncthreads();

  const int wave = t >> 5, lane = t & 31;
  const int n = lane & 15, hi = lane >> 4;
  const int rowBase = blockIdx.x * 128 + wave * 16;

  const float* Arow = E + (size_t)(rowBase + n) * DIM + 2 * hi;  // A: 16x4 f32 layout
  const float* Brow = &lp[n * PSTRIDE + 2 * hi];                 // B: 4x16 f32 layout

  v8f acc0 = {};
  v8f acc1 = {};
  float sqe = 0.0f;
#pragma unroll 4
  for (int k = 0; k < DIM; k += 8) {
    v2f a0 = *(const v2f*)(Arow + k);
    v2f b0 = *(const v2f*)(Brow + k);
    v2f a1 = *(const v2f*)(Arow + k + 4);
    v2f b1 = *(const v2f*)(Brow + k + 4);
    sqe += a0.x * a0.x + a0.y * a0.y + a1.x * a1.x + a1.y * a1.y;
    acc0 = __builtin_amdgcn_wmma_f32_16x16x4_f32(false, a0, false, b0,
                                                 (short)0, acc0, false, false);
    acc1 = __builtin_amdgcn_wmma_f32_16x16x4_f32(false, a1, false, b1,
                                                 (short)0, acc1, false, false);
  }
  // combine the two K-halves of ||e_row||^2 (lanes l and l^16 hold row n=l&15)
  sqe += swz_xor<16>(sqe);
  const float sqp = lsqp[n];

  float wloss = 0.0f;
  float lg[8];
#pragma unroll
  for (int v = 0; v < 8; ++v) {
    const int m = v + 8 * hi;                 // D layout: VGPR v -> row v+8*hi
    const float se = lane_bcast(sqe, m);      // ||e||^2 of output row m
    const float dot = acc0[v] + acc1[v];
    float d2 = se + sqp - 2.0f * dot;
    d2 = fmaxf(d2, 0.0f);
    const float logit = -d2;
    lg[v] = logit;

    // row-wise log-softmax across n (16-lane group, 10 valid cols)
    float mx = (n < NCLS) ? logit : -__builtin_inff();
    mx = fmaxf(mx, swz_xor<1>(mx));
    mx = fmaxf(mx, swz_xor<2>(mx));
    mx = fmaxf(mx, swz_xor<4>(mx));
    mx = fmaxf(mx, swz_xor<8>(mx));
    float ex = (n < NCLS) ? __expf(logit - mx) : 0.0f;
    ex += swz_xor<1>(ex);
    ex += swz_xor<2>(ex);
    ex += swz_xor<4>(ex);
    ex += swz_xor<8>(ex);
    const float lse = __logf(ex);

    const int labv = lab[rowBase + m];
    const float contrib = (n == labv) ? (mx + lse - logit) : 0.0f;
    wloss += contrib;
  }

  // write logits [rows][10]
  if (n < NCLS) {
#pragma unroll
    for (int v = 0; v < 8; ++v) {
      const int m = v + 8 * hi;
      logits[(size_t)(rowBase + m) * NCLS + n] = lg[v];
    }
  }

  // loss: wave -> block -> single atomic
  wloss += swz_xor<1>(wloss);
  wloss += swz_xor<2>(wloss);
  wloss += swz_xor<4>(wloss);
  wloss += swz_xor<8>(wloss);
  wloss += swz_xor<16>(wloss);
  if (lane == 0) lloss[wave] = wloss;
  __syncthreads();
  if (t == 0) {
    float s = 0.0f;
#pragma unroll
    for (int i = 0; i < 8; ++i) s += lloss[i];
    atomicAdd(loss_acc, s);
  }
}

// ---------- kernel 4: finalize loss ----------
__global__ void k_final(const float* __restrict__ ws, float* __restrict__ out) {
  if (threadIdx.x == 0 && blockIdx.x == 0)
    out[0] = ws[OFF_LOSS] * (1.0f / (float)NROWS);
}

extern "C" void kernel_launch(void* const* d_in, const int* in_sizes, int n_in,
                              void* d_out, int out_size, void* d_ws, size_t ws_size,
                              hipStream_t stream) {
  const float* E  = (const float*)d_in[0];
  const int* lab  = (const int*)d_in[1];
  float* out      = (float*)d_out;      // [0] = loss, [1..] = logits
  float* ws       = (float*)d_ws;

  k_init<<<(WS_FLOATS + 255) / 256, 256, 0, stream>>>(ws);
  k_protosum<<<NROWS / 256, 256, 0, stream>>>(E, lab, ws);
  k_proto<<<16, 256, 0, stream>>>(ws);
  k_logits<<<NROWS / 128, 256, 0, stream>>>(E, lab, ws, out + 1, ws + OFF_LOSS);
  k_final<<<1, 32, 0, stream>>>(ws, out);
}